// SemiCurrSinkhornKnopp_stable_15857019256983
// MI455X (gfx1250) — compile-verified
//
#include <hip/hip_runtime.h>
#include <math.h>

// ---------------------------------------------------------------------------
// Semi-relaxed Sinkhorn-Knopp (stable) for MI455X / gfx1250.
//   N=16384 rows, K=1024 cols (+1 slack col), eps=0.1, gamma=1, rho=0.5.
// Hot loop = two L2-resident matvec passes over Q per iteration, both done
// with V_WMMA_F32_16X16X4_F32 tiles using dual accumulator chains for ILP.
// All reductions fixed-order (deterministic).
// ---------------------------------------------------------------------------

#define NROWS   16384
#define KC      1024
#define KP1     1025            // K+1 (real columns)
#define KP      1056            // padded column count (33*32), pad cols are 0
#define NCHUNK  64              // row chunks for Q^T a partials (64*256=16384)
#define NITER   1000

#define EPSV     0.1f
#define INV_EPS  10.0f
#define FI       0.9090909090909091f   // gamma/(gamma+eps)
#define STOPERR_V 1e-10f
#define STABT    1e8f
#define TINYV    1.1920929e-07f        // float32 machine eps
#define PA_V     (1.0f/16384.0f)
#define PB0_V    (0.5f/1024.0f)        // rho/k
#define PBL_V    0.5f                  // 1-rho

typedef __attribute__((ext_vector_type(2))) float v2f;
typedef __attribute__((ext_vector_type(8))) float v8f;

// workspace layout (float offsets)
#define Q_OFF    0L
#define PART_OFF (Q_OFF + (long)NROWS * KP)
#define A_OFF    (PART_OFF + (long)NCHUNK * KP)
#define U_OFF    (A_OFF + NROWS)
#define LSE_OFF  (U_OFF + NROWS)
#define B_OFF    (LSE_OFF + NROWS)
#define LB_OFF   (B_OFF + KP)
#define V_OFF    (LB_OFF + KP)
#define W_OFF    (V_OFF + KP)
#define FLG_OFF  (W_OFF + KC)
// int flags at FLG_OFF: [0]=done [1]=stab_iter [2]=stabled_last [3]=err(f32)

// -------------------------- reductions (wave32) ----------------------------
__device__ __forceinline__ float wred_sum(float v) {
#pragma unroll
  for (int o = 16; o > 0; o >>= 1) v += __shfl_xor(v, o, 32);
  return v;
}
__device__ __forceinline__ float wred_max(float v) {
#pragma unroll
  for (int o = 16; o > 0; o >>= 1) v = fmaxf(v, __shfl_xor(v, o, 32));
  return v;
}
__device__ float block_sum(float v, float* s) {
  int lane = threadIdx.x & 31, wid = threadIdx.x >> 5;
  int nw = (blockDim.x + 31) >> 5;
  v = wred_sum(v);
  __syncthreads();
  if (lane == 0) s[wid] = v;
  __syncthreads();
  float t = (threadIdx.x < nw) ? s[threadIdx.x] : 0.0f;
  if (wid == 0) { t = wred_sum(t); if (lane == 0) s[0] = t; }
  __syncthreads();
  return s[0];
}
__device__ float block_max(float v, float* s) {
  int lane = threadIdx.x & 31, wid = threadIdx.x >> 5;
  int nw = (blockDim.x + 31) >> 5;
  v = wred_max(v);
  __syncthreads();
  if (lane == 0) s[wid] = v;
  __syncthreads();
  float t = (threadIdx.x < nw) ? s[threadIdx.x] : -3.4e38f;
  if (wid == 0) { t = wred_max(t); if (lane == 0) s[0] = t; }
  __syncthreads();
  return s[0];
}

// ------------------ init: lse per row, Q0 = exp(-cost/eps) -----------------
__global__ void k_init_q(const float* __restrict__ logits, float* __restrict__ ws) {
  __shared__ float s[32];
  int i = blockIdx.x;
  const float* x = logits + (long)i * KC;
  float m = -3.4e38f;
  for (int j = threadIdx.x; j < KC; j += blockDim.x) m = fmaxf(m, x[j]);
  m = block_max(m, s);
  float sum = 0.0f;
  for (int j = threadIdx.x; j < KC; j += blockDim.x) sum += expf(x[j] - m);
  sum = block_sum(sum, s);
  float lse = m + logf(sum);             // cost_ij = lse - x_ij  (j<K), 0 at j=K
  if (threadIdx.x == 0) ws[LSE_OFF + i] = lse;
  float* Qr = ws + Q_OFF + (long)i * KP;
  for (int j = threadIdx.x; j < KP; j += blockDim.x) {
    float q;
    if (j < KC)       q = expf((x[j] - lse) * INV_EPS);  // exp(-cost/eps)
    else if (j == KC) q = 1.0f;
    else              q = 0.0f;                          // zero padding
    Qr[j] = q;
  }
}

__global__ void k_init_vec(float* __restrict__ ws) {
  int i = blockIdx.x * blockDim.x + threadIdx.x;
  if (i < NROWS) { ws[A_OFF + i] = 1.0f; ws[U_OFF + i] = 0.0f; }
  if (i < KP) {
    float b0 = (i < KP1) ? (1.0f / (float)KP1) : 0.0f;
    ws[B_OFF + i] = b0; ws[LB_OFF + i] = b0; ws[V_OFF + i] = 0.0f;
  }
  if (i < KC) ws[W_OFF + i] = 1.0f;
  if (i == 0) {
    int* f = (int*)(ws + FLG_OFF);
    f[0] = 0; f[1] = -1; f[2] = 0; ((float*)f)[3] = 1.0f;
  }
}

// ---------------- pass 1: a = Pa / (Q @ b), WMMA f32 16x16x4 ---------------
// One wave owns 16 rows. A-frag = Q[16 rows][4 cols], B-frag = b[4] broadcast
// into all 16 columns -> every column of D holds the 16 row dot-products.
// Dual accumulator chains (even/odd 4-col chunks) break the D->C serial chain.
__global__ void k_matvec_a(float* __restrict__ ws) {
  const int* f = (const int*)(ws + FLG_OFF);
  if (f[0]) return;                              // uniform -> EXEC stays full
  int wid = threadIdx.x >> 5, lane = threadIdx.x & 31;
  int half = lane >> 4, lrow = lane & 15;
  int rb = (blockIdx.x * (blockDim.x >> 5) + wid) * 16;
  const float* qrow = ws + Q_OFF + (long)(rb + lrow) * KP + 2 * half;
  const float* bp   = ws + B_OFF + 2 * half;
  v8f acc0 = {};
  v8f acc1 = {};
#pragma unroll 4
  for (int c = 0; c < KP; c += 8) {
    v2f af0 = *(const v2f*)(qrow + c);           // A[m][k]: Q[rb+lrow][c+2h..]
    v2f bf0 = *(const v2f*)(bp + c);             // B[k][n]=b[c+k] for all n
    v2f af1 = *(const v2f*)(qrow + c + 4);
    v2f bf1 = *(const v2f*)(bp + c + 4);
    acc0 = __builtin_amdgcn_wmma_f32_16x16x4_f32(false, af0, false, bf0,
                                                 (short)0, acc0, false, false);
    acc1 = __builtin_amdgcn_wmma_f32_16x16x4_f32(false, af1, false, bf1,
                                                 (short)0, acc1, false, false);
  }
  if (lrow == 0) {                               // lanes 0 & 16 hold rows 0-7/8-15
    float* a = ws + A_OFF + rb + 8 * half;
#pragma unroll
    for (int r = 0; r < 8; r++) a[r] = PA_V / (acc0[r] + acc1[r]);
  }
}

// -------- pass 2: per-row-chunk partials of c = Q^T a, WMMA 16x16x4 --------
// One wave owns 16 columns x 256 rows. A-frag = a[4] broadcast across rows,
// B-frag = Q[4 rows][16 cols] -> row 0 of D = 16 column partial sums.
__global__ void k_matvec_bT(float* __restrict__ ws) {
  const int* f = (const int*)(ws + FLG_OFF);
  if (f[0]) return;
  int wid = threadIdx.x >> 5, lane = threadIdx.x & 31;
  int half = lane >> 4, lrow = lane & 15;
  int cb = blockIdx.x * 16;
  int chunk = blockIdx.y * (blockDim.x >> 5) + wid;   // 0..63
  int r0 = chunk * 256;
  const float* Q = ws + Q_OFF;
  const float* a = ws + A_OFF;
  v8f acc0 = {};
  v8f acc1 = {};
#pragma unroll 4
  for (int r = r0; r < r0 + 256; r += 8) {
    v2f af0 = *(const v2f*)(a + r + 2 * half);        // A[m][k]=a[r+k], any m
    v2f af1 = *(const v2f*)(a + r + 4 + 2 * half);
    const float* qp0 = Q + (long)(r + 2 * half) * KP + cb + lrow;
    const float* qp1 = qp0 + 4L * KP;
    v2f bf0; bf0.x = qp0[0]; bf0.y = qp0[KP];         // B[k][n]=Q[r+k][cb+n]
    v2f bf1; bf1.x = qp1[0]; bf1.y = qp1[KP];
    acc0 = __builtin_amdgcn_wmma_f32_16x16x4_f32(false, af0, false, bf0,
                                                 (short)0, acc0, false, false);
    acc1 = __builtin_amdgcn_wmma_f32_16x16x4_f32(false, af1, false, bf1,
                                                 (short)0, acc1, false, false);
  }
  if (half == 0)                                      // D row M=0: lanes 0-15
    ws[PART_OFF + (long)chunk * KP + cb + lrow] = acc0[0] + acc1[0];
}

// ------- reduce partials -> b, err, stabilization decision (1 block) -------
__global__ void k_update_b(float* __restrict__ ws, int iter) {
  __shared__ float sred[32];
  __shared__ float sbn[KP1];
  __shared__ int sstab;
  int* f = (int*)(ws + FLG_OFF);
  if (f[0]) return;
  int tid = threadIdx.x;
  float errsq = 0.0f, bmax = 0.0f;
  for (int j = tid; j < KP1; j += blockDim.x) {
    float c = 0.0f;
    for (int p = 0; p < NCHUNK; p++) c += ws[PART_OFF + (long)p * KP + j];
    float bn;
    if (j < KC) bn = powf(PB0_V / c, FI) * ws[W_OFF + j];  // semi-relaxed power
    else        bn = PBL_V / c;
    sbn[j] = bn;
    float d = bn - ws[LB_OFF + j];
    errsq += d * d;
    bmax = fmaxf(bmax, bn);
  }
  errsq = block_sum(errsq, sred);
  bmax  = block_max(bmax, sred);
  float amax = 0.0f;
  for (int i = tid; i < NROWS; i += blockDim.x) amax = fmaxf(amax, ws[A_OFF + i]);
  amax = block_max(amax, sred);
  if (tid == 0) {
    float err = sqrtf(errsq);
    ((float*)f)[3] = err;
    int ns = (fmaxf(amax, bmax) > STABT) ? 1 : 0;
    sstab = ns;
    f[2] = ns;                      // 'stabled' of the last executed body
    if (ns) f[1] = iter;            // stamp enables stab kernels this iter only
    if (err <= STOPERR_V) f[0] = 1; // gates all subsequent iterations
  }
  __syncthreads();
  int ns = sstab;
  for (int j = tid; j < KP1; j += blockDim.x) {
    float bn = sbn[j];
    float nb = ns ? 1.0f : bn;      // stab resets b to ones
    ws[B_OFF + j]  = nb;
    ws[LB_OFF + j] = nb;            // last_b = post-stab b (matches reference)
    if (ns) {
      ws[V_OFF + j] += EPSV * logf(bn + TINYV);
      if (j < KC) ws[W_OFF + j] *= powf(bn, FI - 1.0f);
    }
  }
}

// ----------------------- stabilization: u and Q rebuild --------------------
__global__ void k_stab_u(float* __restrict__ ws, int iter) {
  const int* f = (const int*)(ws + FLG_OFF);
  if (f[1] != iter) return;
  int i = blockIdx.x * blockDim.x + threadIdx.x;
  if (i < NROWS) ws[U_OFF + i] += EPSV * logf(ws[A_OFF + i]);
}

__global__ void k_stab_q(const float* __restrict__ logits, float* __restrict__ ws,
                         int iter) {
  const int* f = (const int*)(ws + FLG_OFF);
  if (f[1] != iter) return;
  int rbase = blockIdx.x * 16;
  for (int rr = 0; rr < 16; rr++) {
    int i = rbase + rr;
    float un = ws[U_OFF + i];
    float li = ws[LSE_OFF + i];
    const float* x = logits + (long)i * KC;
    float* Qr = ws + Q_OFF + (long)i * KP;
    for (int j = threadIdx.x; j < KP; j += blockDim.x) {
      float q;
      if (j < KC)       q = expf((un - (li - x[j]) + ws[V_OFF + j]) * INV_EPS);
      else if (j == KC) q = expf((un + ws[V_OFF + j]) * INV_EPS);
      else              q = 0.0f;
      Qr[j] = q;
    }
  }
}

// ------------------- plan = n * (stabled ? Q : a*Q*b^T), drop slack --------
__global__ void k_final(float* __restrict__ out, const float* __restrict__ ws) {
  const int* f = (const int*)(ws + FLG_OFF);
  int st = f[2];
  int i = blockIdx.x;
  const float* Qr = ws + Q_OFF + (long)i * KP;
  float ai = ws[A_OFF + i];
  for (int j = threadIdx.x; j < KC; j += blockDim.x) {
    float q = Qr[j];
    out[(long)i * KC + j] = 16384.0f * (st ? q : ai * q * ws[B_OFF + j]);
  }
}

// ---------------------------------------------------------------------------
extern "C" void kernel_launch(void* const* d_in, const int* in_sizes, int n_in,
                              void* d_out, int out_size, void* d_ws, size_t ws_size,
                              hipStream_t stream) {
  const float* logits = (const float*)d_in[0];
  float* ws  = (float*)d_ws;
  float* out = (float*)d_out;

  hipLaunchKernelGGL(k_init_q,   dim3(NROWS), dim3(256), 0, stream, logits, ws);
  hipLaunchKernelGGL(k_init_vec, dim3(64),    dim3(256), 0, stream, ws);

  for (int it = 0; it < NITER; ++it) {
    hipLaunchKernelGGL(k_matvec_a,  dim3(128),    dim3(256),  0, stream, ws);
    hipLaunchKernelGGL(k_matvec_bT, dim3(66, 8),  dim3(256),  0, stream, ws);
    hipLaunchKernelGGL(k_update_b,  dim3(1),      dim3(1024), 0, stream, ws, it);
    hipLaunchKernelGGL(k_stab_u,    dim3(16),     dim3(1024), 0, stream, ws, it);
    hipLaunchKernelGGL(k_stab_q,    dim3(1024),   dim3(256),  0, stream, logits, ws, it);
  }

  hipLaunchKernelGGL(k_final, dim3(NROWS), dim3(256), 0, stream, out, ws);
}